// Resize_23433341567035
// MI455X (gfx1250) — compile-verified
//
#include <hip/hip_runtime.h>
#include <cstdint>

// ---------------------------------------------------------------------------
// Trilinear 2x upsample: [2,128,128,128,2] f32 -> [2,256,256,256,2] f32.
// Memory-bound (~300 MB traffic @ 23.3 TB/s ~ 13us). CDNA5 paths used:
//   * global_load_async_to_lds_b128 (ASYNCcnt) to stage input rows in LDS
//   * s_wait_asynccnt 0 before the workgroup barrier
//   * non-temporal stores for the write-only 268 MB output stream
// Zoom=2.0 => weights are {1, 1/2}; clamped-neighbor averaging reproduces the
// reference's clip-then-floor edge behavior exactly (0.5*(v+v) == v).
// ---------------------------------------------------------------------------

#define BB 2
#define DD 128
#define HH 128
#define WW 128
#define CC 2

#define IN_H_STR   (WW * CC)            /* 256      */
#define IN_D_STR   (HH * WW * CC)       /* 32768    */
#define IN_B_STR   (DD * HH * WW * CC)  /* 4194304  */

#define OUT_H_STR  (2 * WW * CC)              /* 512      */
#define OUT_D_STR  (2 * HH * 2 * WW * CC)     /* 131072   */
#define OUT_B_STR  (2 * DD * OUT_D_STR)       /* 33554432 */

typedef __attribute__((ext_vector_type(4))) float v4f;  // legal for NT builtins

__device__ __forceinline__ void async_load16(uint32_t lds_byte_addr, const float* g) {
  // CDNA5 async VMEM->LDS copy, tracked by ASYNCcnt (no VGPR data return).
  asm volatile("global_load_async_to_lds_b128 %0, %1, off"
               :: "v"(lds_byte_addr), "v"(g)
               : "memory");
}

__device__ __forceinline__ void wait_async0() {
#if __has_builtin(__builtin_amdgcn_s_wait_asynccnt)
  __builtin_amdgcn_s_wait_asynccnt(0);
#else
  asm volatile("s_wait_asynccnt 0" ::: "memory");
#endif
}

__device__ __forceinline__ float2 f2avg(float2 a, float2 b) {
  return make_float2(0.5f * (a.x + b.x), 0.5f * (a.y + b.y));
}
__device__ __forceinline__ float2 f2avg4(float2 a, float2 b, float2 c, float2 d) {
  return make_float2(0.25f * (a.x + b.x + c.x + d.x),
                     0.25f * (a.y + b.y + c.y + d.y));
}
__device__ __forceinline__ void store_pair_nt(float* p, float2 a, float2 b) {
  // {w_even(c0), w_even(c1), w_odd(c0), w_odd(c1)} -> one coalesced b128 store
  v4f o;
  o.x = a.x;
  o.y = a.y;
  o.z = 0.5f * (a.x + b.x);
  o.w = 0.5f * (a.y + b.y);
  __builtin_nontemporal_store(o, reinterpret_cast<v4f*>(p));
}

__global__ __launch_bounds__(128)
void resize2x_trilinear_kernel(const float* __restrict__ vol, float* __restrict__ out) {
  // 4 input rows (d,h neighborhood), 128 voxels * 2 ch each = 4 KB.
  __shared__ alignas(16) float smem[4 * WW * CC];

  const int tid = threadIdx.x;        // k index, 0..127
  const int bid = blockIdx.x;         // 0..32767
  const int j = bid & (HH - 1);
  const int i = (bid >> 7) & (DD - 1);
  const int b = bid >> 14;

  const int i1 = min(i + 1, DD - 1);
  const int j1 = min(j + 1, HH - 1);

  const uint32_t lds_base = (uint32_t)(uintptr_t)(&smem[0]);
  const float* vbase = vol + (size_t)b * IN_B_STR;

  // Stage 4 rows: 256 x 16B async chunks, 2 per thread, all lanes active.
  #pragma unroll
  for (int it = tid; it < 256; it += 128) {
    const int r = it >> 6;        // row 0..3  (bit1 = di, bit0 = dj)
    const int q = it & 63;        // 16B chunk within row
    const int ci = (r & 2) ? i1 : i;
    const int cj = (r & 1) ? j1 : j;
    const float* g = vbase + (size_t)ci * IN_D_STR + (size_t)cj * IN_H_STR + q * 4;
    async_load16(lds_base + (uint32_t)(r * 1024 + q * 16), g);
  }
  wait_async0();
  __syncthreads();

  const int k = tid;
  const float* r0 = smem + 0 * IN_H_STR + 2 * k;  // (i , j )
  const float* r1 = smem + 1 * IN_H_STR + 2 * k;  // (i , j1)
  const float* r2 = smem + 2 * IN_H_STR + 2 * k;  // (i1, j )
  const float* r3 = smem + 3 * IN_H_STR + 2 * k;  // (i1, j1)

  // A = voxel k (both channels), B = voxel k+1 (clamped at the edge).
  float2 A00 = *(const float2*)r0;
  float2 A01 = *(const float2*)r1;
  float2 A10 = *(const float2*)r2;
  float2 A11 = *(const float2*)r3;
  float2 B00, B01, B10, B11;
  if (k < WW - 1) {
    B00 = *(const float2*)(r0 + 2);
    B01 = *(const float2*)(r1 + 2);
    B10 = *(const float2*)(r2 + 2);
    B11 = *(const float2*)(r3 + 2);
  } else {
    B00 = A00; B01 = A01; B10 = A10; B11 = A11;
  }

  // Bilinear combos over (d,h) for the even-w and odd-w contributions.
  float2 PA00 = A00;
  float2 PA01 = f2avg(A00, A01);
  float2 PA10 = f2avg(A00, A10);
  float2 PA11 = f2avg4(A00, A01, A10, A11);
  float2 PB00 = B00;
  float2 PB01 = f2avg(B00, B01);
  float2 PB10 = f2avg(B00, B10);
  float2 PB11 = f2avg4(B00, B01, B10, B11);

  float* obase = out + (size_t)b * OUT_B_STR
                     + (size_t)(2 * i) * OUT_D_STR
                     + (size_t)(2 * j) * OUT_H_STR
                     + (size_t)k * 4;

  store_pair_nt(obase,                         PA00, PB00);  // (2i  , 2j  )
  store_pair_nt(obase + OUT_H_STR,             PA01, PB01);  // (2i  , 2j+1)
  store_pair_nt(obase + OUT_D_STR,             PA10, PB10);  // (2i+1, 2j  )
  store_pair_nt(obase + OUT_D_STR + OUT_H_STR, PA11, PB11);  // (2i+1, 2j+1)
}

extern "C" void kernel_launch(void* const* d_in, const int* in_sizes, int n_in,
                              void* d_out, int out_size, void* d_ws, size_t ws_size,
                              hipStream_t stream) {
  (void)in_sizes; (void)n_in; (void)d_ws; (void)ws_size; (void)out_size;
  const float* vol = (const float*)d_in[0];
  float* out = (float*)d_out;
  const int nblocks = BB * DD * HH;  // 32768 blocks, one per (b, d_in, h_in)
  resize2x_trilinear_kernel<<<nblocks, 128, 0, stream>>>(vol, out);
}